// GATConvModel_17995912970867
// MI455X (gfx1250) — compile-verified
//
#include <hip/hip_runtime.h>
#include <math.h>

// ---------------------------------------------------------------------------
// GATConv x6 for MI455X (gfx1250, wave32).  GEMMs run through
// V_WMMA_F32_16X16X4_F32 (full fp32 matrix pipe -> exact precision vs ref).
// W is pre-transposed to zero-padded column-major; each wave computes a 16x64
// C slab.  Inner loop is software-pipelined: next step's 5 fragments are
// loaded while the current 4 WMMAs execute.  Edge softmax/scatter via
// hardware f32 atomics.
// ---------------------------------------------------------------------------

typedef float v2f __attribute__((ext_vector_type(2)));
typedef float v8f __attribute__((ext_vector_type(8)));

#define NB    32                    // batch (graphs)
#define NPER  512                   // nodes per graph
#define NTOT  (NB * NPER)           // 16384 nodes total
#define NEDGE 4096                  // edges per graph
#define NE_TOT (NB * NEDGE + NTOT)  // 147456 edges incl. self loops
#define NEG_SLOPE 0.2f

#define WMMA_F32(a, b, c) __builtin_amdgcn_wmma_f32_16x16x4_f32( \
        false, (a), false, (b), (short)0, (c), false, false)

// float atomic max via monotonic integer mapping (valid for mixed signs):
// non-negative floats compare correctly as signed ints (max), negative floats
// compare inversely as unsigned ints (min).  Init value must be -inf.
static __device__ __forceinline__ void atomicMaxF32(float* addr, float val) {
    if (val >= 0.0f) atomicMax((int*)addr, __float_as_int(val));
    else             atomicMin((unsigned int*)addr, (unsigned int)__float_as_int(val));
}

// ---------------- edge-index preprocessing --------------------------------
__global__ void k_max_offset(const int* __restrict__ ei, int n, int* __restrict__ off) {
    __shared__ int sm[256];
    int tid = threadIdx.x;
    int v = 0;
    for (int i = tid; i < n; i += blockDim.x) v = max(v, ei[i]);
    sm[tid] = v; __syncthreads();
    for (int s = blockDim.x >> 1; s > 0; s >>= 1) {
        if (tid < s) sm[tid] = max(sm[tid], sm[tid + s]);
        __syncthreads();
    }
    if (tid == 0) *off = sm[0] + 1;
}

__global__ void k_build_edges(const int* __restrict__ ei, const int* __restrict__ offp,
                              int* __restrict__ src, int* __restrict__ dst) {
    int i = blockIdx.x * blockDim.x + threadIdx.x;
    if (i >= NE_TOT) return;
    int off = *offp;
    if (i < NB * NEDGE) {
        int b = i / NEDGE, j = i - b * NEDGE;
        src[i] = ei[j]         + off * b;   // row 0 = src
        dst[i] = ei[NEDGE + j] + off * b;   // row 1 = dst
    } else {
        int k = i - NB * NEDGE;             // self loops
        src[i] = k; dst[i] = k;
    }
}

// ---------------- W pre-transpose: Wt[Npad][K] = W[K][N]^T, zero padded ----
__global__ void k_prepW(const float* __restrict__ W, float* __restrict__ Wt,
                        int K, int N, int Npad) {
    int i = blockIdx.x * blockDim.x + threadIdx.x;
    if (i >= Npad * K) return;
    int c = i / K, k = i - c * K;
    Wt[i] = (c < N) ? W[(size_t)k * N + c] : 0.0f;
}

// ---------------- fp32 WMMA GEMM:  H[M,N] = X[M,K] * Wt[N,K]^T ------------
// One wave per 16(M) x 64(N) slab: 4 accumulators share one A fragment.
// K stepped by 4 through V_WMMA_F32_16X16X4_F32, software-pipelined so the
// next step's fragments are in flight while the current WMMAs execute.
__global__ __launch_bounds__(256) void k_gemm_wmma_f32(
        const float* __restrict__ X, const float* __restrict__ Wt,
        float* __restrict__ H, int M, int K, int N, int tiles, int tilesN) {
    int lane = threadIdx.x & 31;
    int wave = threadIdx.x >> 5;
    int tile = blockIdx.x * (blockDim.x >> 5) + wave;
    if (tile >= tiles) return;                 // whole-wave exit: EXEC stays all-1s
    int tm = tile / tilesN, tn = tile - tm * tilesN;
    int half = lane >> 4;                      // 0: K k,k+1   1: K k+2,k+3
    int l    = lane & 15;
    int row  = tm * 16 + l;                    // A row for this lane
    int cb   = tn * 64 + l;                    // base column for this lane

    const float* xr  = X  + (size_t)row * K        + half * 2;   // A stream
    const float* wt0 = Wt + (size_t)(cb +  0) * K  + half * 2;   // B streams
    const float* wt1 = Wt + (size_t)(cb + 16) * K  + half * 2;
    const float* wt2 = Wt + (size_t)(cb + 32) * K  + half * 2;
    const float* wt3 = Wt + (size_t)(cb + 48) * K  + half * 2;

    v8f acc0 = {}, acc1 = {}, acc2 = {}, acc3 = {};

    // prologue: load first fragments
    v2f a  = *(const v2f*)xr;
    v2f b0 = *(const v2f*)wt0;
    v2f b1 = *(const v2f*)wt1;
    v2f b2 = *(const v2f*)wt2;
    v2f b3 = *(const v2f*)wt3;

    // pipelined steady state: issue next loads, then consume current fragments
    for (int k = 4; k < K; k += 4) {
        xr += 4; wt0 += 4; wt1 += 4; wt2 += 4; wt3 += 4;
        v2f an  = *(const v2f*)xr;
        v2f bn0 = *(const v2f*)wt0;
        v2f bn1 = *(const v2f*)wt1;
        v2f bn2 = *(const v2f*)wt2;
        v2f bn3 = *(const v2f*)wt3;
        acc0 = WMMA_F32(a, b0, acc0);
        acc1 = WMMA_F32(a, b1, acc1);
        acc2 = WMMA_F32(a, b2, acc2);
        acc3 = WMMA_F32(a, b3, acc3);
        a = an; b0 = bn0; b1 = bn1; b2 = bn2; b3 = bn3;
    }
    // epilogue: consume last fragments
    acc0 = WMMA_F32(a, b0, acc0);
    acc1 = WMMA_F32(a, b1, acc1);
    acc2 = WMMA_F32(a, b2, acc2);
    acc3 = WMMA_F32(a, b3, acc3);

    // C/D layout: VGPR r -> row (tm*16 + half*8 + r), col = tn*64 + j*16 + l
    int orow = tm * 16 + half * 8;
    v8f accs[4] = {acc0, acc1, acc2, acc3};
    #pragma unroll
    for (int j = 0; j < 4; ++j) {
        int c = cb + j * 16;
        if (c < N) {
            #pragma unroll
            for (int r = 0; r < 8; ++r)
                H[(size_t)(orow + r) * N + c] = accs[j][r];
        }
    }
}

// ---------------- per-node attention logits (wave per node) ---------------
__global__ void k_logits(const float* __restrict__ H, const float* __restrict__ a_s,
                         const float* __restrict__ a_d, float* __restrict__ s,
                         float* __restrict__ t, int n, int dout) {
    int wid  = (blockIdx.x * blockDim.x + threadIdx.x) >> 5;
    int lane = threadIdx.x & 31;
    if (wid >= n) return;
    const float* h = H + (size_t)wid * dout;
    float ss = 0.0f, tt = 0.0f;
    for (int f = lane; f < dout; f += 32) {
        float v = h[f];
        ss += v * a_s[f];
        tt += v * a_d[f];
    }
    #pragma unroll
    for (int o = 16; o > 0; o >>= 1) {
        ss += __shfl_down(ss, o, 32);
        tt += __shfl_down(tt, o, 32);
    }
    if (lane == 0) { s[wid] = ss; t[wid] = tt; }
}

// ---------------- per-layer init: zero agg, m=-inf, denom=0 ---------------
__global__ void k_init(float* __restrict__ agg, float* __restrict__ m,
                       float* __restrict__ denom, int n, int dout) {
    int i = blockIdx.x * blockDim.x + threadIdx.x;
    if (i < n * dout) agg[i] = 0.0f;
    if (i < n) { m[i] = -__builtin_inff(); denom[i] = 0.0f; }
}

// ---------------- edge pass A: leaky-relu logit + segment max -------------
__global__ void k_edgeA(const int* __restrict__ src, const int* __restrict__ dst,
                        const float* __restrict__ s, const float* __restrict__ t,
                        float* __restrict__ e, float* __restrict__ m) {
    int i = blockIdx.x * blockDim.x + threadIdx.x;
    if (i >= NE_TOT) return;
    float v = s[src[i]] + t[dst[i]];
    v = (v >= 0.0f) ? v : NEG_SLOPE * v;
    e[i] = v;
    atomicMaxF32(&m[dst[i]], v);
}

// ---------------- edge pass B: exp + segment sum --------------------------
__global__ void k_edgeB(const int* __restrict__ dst, const float* __restrict__ e,
                        const float* __restrict__ m, float* __restrict__ ex,
                        float* __restrict__ denom) {
    int i = blockIdx.x * blockDim.x + threadIdx.x;
    if (i >= NE_TOT) return;
    int d = dst[i];
    float v = expf(e[i] - m[d]);
    ex[i] = v;
    atomicAdd(&denom[d], v);
}

// ---------------- edge pass C: weighted scatter-add (wave per edge) -------
__global__ void k_edgeC(const int* __restrict__ src, const int* __restrict__ dst,
                        const float* __restrict__ ex, const float* __restrict__ denom,
                        const float* __restrict__ H, float* __restrict__ agg, int dout) {
    int eid  = (blockIdx.x * blockDim.x + threadIdx.x) >> 5;
    int lane = threadIdx.x & 31;
    if (eid >= NE_TOT) return;
    int sN = src[eid], dN = dst[eid];
    float alpha = ex[eid] / denom[dN];
    const float* hs = H + (size_t)sN * dout;
    float* ag = agg + (size_t)dN * dout;
    for (int f = lane; f < dout; f += 32)
        atomicAdd(&ag[f], alpha * hs[f]);
}

// ---------------- epilogue: bias (+ optional relu) ------------------------
__global__ void k_epilogue(const float* __restrict__ agg, const float* __restrict__ b,
                           float* __restrict__ out, int n, int dout, int do_relu) {
    int i = blockIdx.x * blockDim.x + threadIdx.x;
    if (i >= n * dout) return;
    int f = i % dout;
    float v = agg[i] + b[f];
    if (do_relu) v = fmaxf(v, 0.0f);
    out[i] = v;
}

// ---------------------------------------------------------------------------
extern "C" void kernel_launch(void* const* d_in, const int* in_sizes, int n_in,
                              void* d_out, int out_size, void* d_ws, size_t ws_size,
                              hipStream_t stream) {
    const float* points = (const float*)d_in[0];
    const int*   ei     = (const int*)d_in[1];
    const float *W[6], *As[6], *Ad[6], *Bb[6];
    for (int l = 0; l < 6; ++l) {
        W[l]  = (const float*)d_in[2 + 4 * l];
        As[l] = (const float*)d_in[3 + 4 * l];
        Ad[l] = (const float*)d_in[4 + 4 * l];
        Bb[l] = (const float*)d_in[5 + 4 * l];
    }
    static const int dims[7] = {32, 64, 256, 1024, 256, 64, 3};

    // ---- workspace carve-up (all 256B aligned) ----
    char* ws = (char*)d_ws;
    const size_t big = (size_t)NTOT * 1024 * sizeof(float);   // 64 MB
    float* P0 = (float*)ws; ws += big;
    float* P1 = (float*)ws; ws += big;
    float* P2 = (float*)ws; ws += big;
    float* Wtb   = (float*)ws; ws += (size_t)1024 * 1024 * sizeof(float); // 4 MB
    float* sbuf  = (float*)ws; ws += (size_t)NTOT * sizeof(float);
    float* tbuf  = (float*)ws; ws += (size_t)NTOT * sizeof(float);
    float* mbuf  = (float*)ws; ws += (size_t)NTOT * sizeof(float);
    float* dbuf  = (float*)ws; ws += (size_t)NTOT * sizeof(float);
    float* ebuf  = (float*)ws; ws += (size_t)NE_TOT * sizeof(float);
    float* exbuf = (float*)ws; ws += (size_t)NE_TOT * sizeof(float);
    int*   srcb  = (int*)ws;   ws += (size_t)NE_TOT * sizeof(int);
    int*   dstb  = (int*)ws;   ws += (size_t)NE_TOT * sizeof(int);
    int*   offb  = (int*)ws;   ws += 256;

    // ---- build batched edge list (offset = max(edge_index)+1, + self loops)
    k_max_offset<<<1, 256, 0, stream>>>(ei, 2 * NEDGE, offb);
    k_build_edges<<<(NE_TOT + 255) / 256, 256, 0, stream>>>(ei, offb, srcb, dstb);

    // ---- 6 GATConv layers, rotating feature buffers ----
    float* bufs[3] = {P0, P1, P2};
    const float* x = points;
    int rot = 0;
    for (int l = 0; l < 6; ++l) {
        int K = dims[l], N = dims[l + 1];
        float* h   = bufs[rot];
        float* agg = bufs[(rot + 1) % 3];

        int tilesN = (N + 63) / 64;                 // 16x64 slabs per wave
        int Npad   = tilesN * 64;
        int tiles  = (NTOT / 16) * tilesN;
        k_prepW<<<(Npad * K + 255) / 256, 256, 0, stream>>>(W[l], Wtb, K, N, Npad);
        k_gemm_wmma_f32<<<(tiles + 7) / 8, 256, 0, stream>>>(x, Wtb, h, NTOT, K, N,
                                                             tiles, tilesN);
        k_init<<<(NTOT * N + 255) / 256, 256, 0, stream>>>(agg, mbuf, dbuf, NTOT, N);
        k_logits<<<(NTOT * 32 + 255) / 256, 256, 0, stream>>>(h, As[l], Ad[l],
                                                              sbuf, tbuf, NTOT, N);
        k_edgeA<<<(NE_TOT + 255) / 256, 256, 0, stream>>>(srcb, dstb, sbuf, tbuf,
                                                          ebuf, mbuf);
        k_edgeB<<<(NE_TOT + 255) / 256, 256, 0, stream>>>(dstb, ebuf, mbuf,
                                                          exbuf, dbuf);
        k_edgeC<<<(NE_TOT * 32 + 255) / 256, 256, 0, stream>>>(srcb, dstb, exbuf,
                                                               dbuf, h, agg, N);
        float* outp = (l == 5) ? (float*)d_out : agg;   // in-place act for 0..4
        k_epilogue<<<(NTOT * N + 255) / 256, 256, 0, stream>>>(agg, Bb[l], outp,
                                                               NTOT, N, l < 5);
        x = outp;                 // next layer input (agg holds activations)
        rot = (rot + 2) % 3;
    }
}